// PlainGCN_14491219657415
// MI455X (gfx1250) — compile-verified
//
#include <hip/hip_runtime.h>

#define N_NODES   100000
#define N_EDGES   800000
#define F_IN      3
#define HIDDEN    256
#define N_CLASSES 2
#define N_GRAPHS  64

typedef float v2f __attribute__((ext_vector_type(2)));
typedef float v8f __attribute__((ext_vector_type(8)));

// ---------------- degree ----------------
__global__ void k_init_deg(float* deg) {
    int i = blockIdx.x * blockDim.x + threadIdx.x;
    if (i < N_NODES) deg[i] = 1.0f;                      // self-loop
}

__global__ void k_count_deg(const int* __restrict__ dst, float* deg) {
    int e = blockIdx.x * blockDim.x + threadIdx.x;
    if (e < N_EDGES) atomicAdd(&deg[dst[e]], 1.0f);
}

__global__ void k_finalize_deg(const float* __restrict__ deg,
                               float* __restrict__ dis, float* __restrict__ di) {
    int i = blockIdx.x * blockDim.x + threadIdx.x;
    if (i < N_NODES) {
        float d = deg[i];
        dis[i] = rsqrtf(d);
        di[i]  = 1.0f / d;
    }
}

// ---------------- layer 1: [N,3] @ [3,256] ----------------
__global__ __launch_bounds__(256) void k_gemm_l1(const float* __restrict__ x,
                                                 const float* __restrict__ W,
                                                 float* __restrict__ out) {
    int idx = blockIdx.x * 256 + threadIdx.x;            // over N*256 exactly
    int n = idx >> 8, c = idx & 255;
    float x0 = x[n * 3 + 0], x1 = x[n * 3 + 1], x2 = x[n * 3 + 2];
    out[idx] = fmaf(x0, W[c], fmaf(x1, W[256 + c], x2 * W[512 + c]));
}

// ---------------- layers 2..5: [N,256] @ [256,256] via fp32 WMMA ----------------
// Block = 8 waves, each wave computes a 16x64 output tile.
// B column slice (256x64) staged in LDS in k-PAIR-INTERLEAVED layout:
//   Bs[p*128 + c*2 + (k&1)]  where p = k>>1, c = column within the 64-slice.
// Each WMMA B fragment {B[k2][n], B[k2+1][n]} is then ONE aligned ds_load_b64,
// landing directly in an even register pair (no repack movs before v_wmma).
__global__ __launch_bounds__(256) void k_gemm_wmma(const float* __restrict__ A,
                                                   const float* __restrict__ B,
                                                   float* __restrict__ C) {
    __shared__ float Bs[HIDDEN * 64];                    // 64 KB, pair-interleaved
    const int cg      = blockIdx.x & 3;                  // column group: 4 x 64
    const int rg      = blockIdx.x >> 2;                 // row group: 8 tiles of 16
    const int colBase = cg * 64;

    // cooperative fill: 8192 (k-pair, col) entries, 32 per thread
    for (int i = 0; i < 32; ++i) {
        int lin = i * 256 + threadIdx.x;                 // [0, 8192)
        int p   = lin >> 6;                              // k pair index 0..127
        int c   = lin & 63;                              // column 0..63
        v2f v;
        v.x = B[(2 * p    ) * HIDDEN + colBase + c];
        v.y = B[(2 * p + 1) * HIDDEN + colBase + c];
        *(v2f*)(&Bs[p * 128 + c * 2]) = v;               // one ds_store_b64
    }
    __syncthreads();

    const int wave    = threadIdx.x >> 5;
    const int lane    = threadIdx.x & 31;
    const int rowTile = rg * 8 + wave;
    if (rowTile >= N_NODES / 16) return;                 // N_NODES % 16 == 0

    const int m   = lane & 15;                           // row of A tile / col of B tile
    const int kh  = lane >> 4;                           // k-half select
    const int row0 = rowTile * 16;
    const float* Arow = A + (size_t)(row0 + m) * HIDDEN + 2 * kh;
    const float* BsLane = &Bs[m * 2];                    // + p*128 + j*32 below

    v8f acc0 = {}, acc1 = {}, acc2 = {}, acc3 = {};
#pragma unroll 4
    for (int kb = 0; kb < HIDDEN; kb += 4) {
        v2f a = *(const v2f*)(Arow + kb);                // {A[m][k2], A[m][k2+1]}
        int p = (kb >> 1) + kh;                          // k2/2
        const float* bp = BsLane + p * 128;
        v2f b0 = *(const v2f*)(bp);                      // {B[k2][n], B[k2+1][n]}
        v2f b1 = *(const v2f*)(bp + 32);
        v2f b2 = *(const v2f*)(bp + 64);
        v2f b3 = *(const v2f*)(bp + 96);
        acc0 = __builtin_amdgcn_wmma_f32_16x16x4_f32(false, a, false, b0, (short)0, acc0, false, false);
        acc1 = __builtin_amdgcn_wmma_f32_16x16x4_f32(false, a, false, b1, (short)0, acc1, false, false);
        acc2 = __builtin_amdgcn_wmma_f32_16x16x4_f32(false, a, false, b2, (short)0, acc2, false, false);
        acc3 = __builtin_amdgcn_wmma_f32_16x16x4_f32(false, a, false, b3, (short)0, acc3, false, false);
    }

    // D layout: VGPR r -> row = row0 + r + 8*kh, col = colBase + j*16 + (lane&15)
    for (int r = 0; r < 8; ++r) {
        int row = row0 + r + 8 * kh;
        float* Crow = C + (size_t)row * HIDDEN + colBase + m;
        Crow[0]  = acc0[r];
        Crow[16] = acc1[r];
        Crow[32] = acc2[r];
        Crow[48] = acc3[r];
    }
}

// ---------------- aggregation ----------------
// agg = deg_inv * (hW) + bias   (self-loop + bias seed)
__global__ __launch_bounds__(256) void k_seed_agg(const float* __restrict__ tmp,
                                                  const float* __restrict__ di,
                                                  const float* __restrict__ bias,
                                                  float* __restrict__ agg) {
    int idx = blockIdx.x * 256 + threadIdx.x;
    int n = idx >> 8, c = idx & 255;
    agg[idx] = fmaf(di[n], tmp[idx], bias[c]);
}

// agg[dst] += dis[src]*dis[dst] * tmp[src]  -- L2-resident f32 atomics
__global__ __launch_bounds__(256) void k_edge_scatter(const int* __restrict__ src,
                                                      const int* __restrict__ dst,
                                                      const float* __restrict__ dis,
                                                      const float* __restrict__ tmp,
                                                      float* __restrict__ agg) {
    int t = threadIdx.x;
    int e = blockIdx.x * 4 + (t >> 6);                   // 4 edges / block, 64 thr each
    if (e >= N_EDGES) return;
    int s = src[e], d = dst[e];
    float norm = dis[s] * dis[d];
    int c = (t & 63) * 4;
    const float4 v = *(const float4*)(&tmp[(size_t)s * HIDDEN + c]);
    float* ap = &agg[(size_t)d * HIDDEN + c];
    atomicAdd(ap + 0, norm * v.x);
    atomicAdd(ap + 1, norm * v.y);
    atomicAdd(ap + 2, norm * v.z);
    atomicAdd(ap + 3, norm * v.w);
}

__global__ __launch_bounds__(256) void k_relu_ip(float* h) {
    int idx = blockIdx.x * 256 + threadIdx.x;
    h[idx] = fmaxf(h[idx], 0.0f);
}

// ---------------- pooling + head ----------------
__global__ void k_zero(float* p, int n) {
    int i = blockIdx.x * blockDim.x + threadIdx.x;
    if (i < n) p[i] = 0.0f;
}

__global__ void k_pool_count(const int* __restrict__ batch, float* counts) {
    int i = blockIdx.x * blockDim.x + threadIdx.x;
    if (i < N_NODES) atomicAdd(&counts[batch[i]], 1.0f);
}

__global__ __launch_bounds__(256) void k_pool_acc(const float* __restrict__ h,
                                                  const int* __restrict__ batch,
                                                  float* __restrict__ pooled) {
    int idx = blockIdx.x * 256 + threadIdx.x;
    int n = idx >> 8, c = idx & 255;
    atomicAdd(&pooled[batch[n] * HIDDEN + c], h[idx]);
}

__global__ void k_out_head(const float* __restrict__ pooled,
                           const float* __restrict__ counts,
                           const float* __restrict__ Wout,
                           const float* __restrict__ bout,
                           float* __restrict__ out) {
    int t = threadIdx.x;
    if (t >= N_GRAPHS * N_CLASSES) return;
    int g = t >> 1, k = t & 1;
    float s = 0.0f;
    for (int c = 0; c < HIDDEN; ++c)
        s += pooled[g * HIDDEN + c] * Wout[c * N_CLASSES + k];
    out[t] = s / fmaxf(counts[g], 1.0f) + bout[k];
}

extern "C" void kernel_launch(void* const* d_in, const int* in_sizes, int n_in,
                              void* d_out, int out_size, void* d_ws, size_t ws_size,
                              hipStream_t stream) {
    (void)in_sizes; (void)n_in; (void)out_size; (void)ws_size;
    const float* x     = (const float*)d_in[0];
    const int*   edge  = (const int*)d_in[1];
    const int*   srcE  = edge;                  // edge_index[0]
    const int*   dstE  = edge + N_EDGES;        // edge_index[1]
    const int*   batch = (const int*)d_in[2];
    const float* W[5]  = {(const float*)d_in[3],  (const float*)d_in[5],
                          (const float*)d_in[7],  (const float*)d_in[9],
                          (const float*)d_in[11]};
    const float* Bv[5] = {(const float*)d_in[4],  (const float*)d_in[6],
                          (const float*)d_in[8],  (const float*)d_in[10],
                          (const float*)d_in[12]};
    const float* Wout  = (const float*)d_in[13];
    const float* bout  = (const float*)d_in[14];
    float* out = (float*)d_out;

    // workspace layout (floats)
    float* ws = (float*)d_ws;
    size_t off = 0;
    float* deg    = ws + off; off += 100096;
    float* dis    = ws + off; off += 100096;
    float* di     = ws + off; off += 100096;
    float* counts = ws + off; off += 256;                 // 64 used
    float* pooled = ws + off; off += (size_t)N_GRAPHS * HIDDEN;
    float* buf0   = ws + off; off += (size_t)N_NODES * HIDDEN;
    float* buf1   = ws + off; off += (size_t)N_NODES * HIDDEN;
    float* buf2   = ws + off; off += (size_t)N_NODES * HIDDEN;

    const int NB         = (N_NODES * HIDDEN) / 256;      // 100000 blocks
    const int gemmBlocks = ((N_NODES / 16 + 7) / 8) * 4;  // 782 row-groups x 4 col-groups

    // degrees
    k_init_deg<<<(N_NODES + 255) / 256, 256, 0, stream>>>(deg);
    k_count_deg<<<(N_EDGES + 255) / 256, 256, 0, stream>>>(dstE, deg);
    k_finalize_deg<<<(N_NODES + 255) / 256, 256, 0, stream>>>(deg, dis, di);

    // layer 1 (K=3, VALU gemm)
    float* tmp = buf0;
    float* agg = buf1;
    k_gemm_l1<<<NB, 256, 0, stream>>>(x, W[0], tmp);
    k_seed_agg<<<NB, 256, 0, stream>>>(tmp, di, Bv[0], agg);
    k_edge_scatter<<<N_EDGES / 4, 256, 0, stream>>>(srcE, dstE, dis, tmp, agg);
    k_relu_ip<<<NB, 256, 0, stream>>>(agg);
    float* h  = agg;     // buf1
    float* f0 = buf0;    // free
    float* f1 = buf2;    // free

    // layers 2..5 (WMMA gemm)
    for (int l = 1; l < 5; ++l) {
        tmp = f0;
        agg = f1;
        k_gemm_wmma<<<gemmBlocks, 256, 0, stream>>>(h, W[l], tmp);
        k_seed_agg<<<NB, 256, 0, stream>>>(tmp, di, Bv[l], agg);
        k_edge_scatter<<<N_EDGES / 4, 256, 0, stream>>>(srcE, dstE, dis, tmp, agg);
        k_relu_ip<<<NB, 256, 0, stream>>>(agg);
        f0 = h; f1 = tmp; h = agg;
    }

    // pooling + head (counts and pooled are contiguous: zero both at once)
    k_zero<<<(256 + N_GRAPHS * HIDDEN + 255) / 256, 256, 0, stream>>>(counts, 256 + N_GRAPHS * HIDDEN);
    k_pool_count<<<(N_NODES + 255) / 256, 256, 0, stream>>>(batch, counts);
    k_pool_acc<<<NB, 256, 0, stream>>>(h, batch, pooled);
    k_out_head<<<1, 128, 0, stream>>>(pooled, counts, Wout, bout, out);
}